// BrainGAE_model_44624710205922
// MI455X (gfx1250) — compile-verified
//
#include <hip/hip_runtime.h>
#include <math.h>

typedef float v2f __attribute__((ext_vector_type(2)));
typedef float v8f __attribute__((ext_vector_type(8)));

namespace {

constexpr int GB  = 256;   // batch
constexpr int GN  = 200;   // nodes
constexpr int GH  = 128;   // HID
constexpr int GD  = 64;    // DIM
constexpr int GK1 = 100;
constexpr int GK2 = 50;
constexpr long NN  = (long)GN * GN;     // 40000
constexpr long BNN = (long)GB * NN;     // 10,240,000

// ---- d_out layout (floats), tuple order: xy, sig(s1), sig(s2), adj_org, adj_logits
constexpr long XY_OFF   = 0;
constexpr long S1_OFF   = XY_OFF + (long)GB * 2;
constexpr long S2_OFF   = S1_OFF + (long)GB * GK1;
constexpr long AORG_OFF = S2_OFF + (long)GB * GK2;
constexpr long ALOG_OFF = AORG_OFF + BNN;

// ---- workspace layout (floats)
constexpr long W_AN   = 0;
constexpr long W_T0   = W_AN  + BNN;                  // HM/HS, later adj_s
constexpr long W_T1   = W_T0  + BNN;                  // m -> z (in place)
constexpr long W_T2   = W_T1  + BNN;                  // s
constexpr long W_XB   = W_T2  + BNN;                  // [B,N,HID]
constexpr long W_H    = W_XB  + (long)GB*GN*GH;       // [B,N,HID]
constexpr long W_XW1  = W_H   + (long)GB*GN*GH;       // [B,N,DIM]
constexpr long W_X1   = W_XW1 + (long)GB*GN*GD;       // conv1 out [B,N,DIM]
constexpr long W_XS1  = W_X1  + (long)GB*GN*GD;       // [B,K1,DIM]
constexpr long W_AS1  = W_XS1 + (long)GB*GK1*GD;      // [B,K1,K1]
constexpr long W_XW2  = W_AS1 + (long)GB*GK1*GK1;     // [B,K1,DIM]
constexpr long W_X2   = W_XW2 + (long)GB*GK1*GD;      // conv2 out [B,K1,DIM]
constexpr long W_XS2  = W_X2  + (long)GB*GK1*GD;      // [B,K2,DIM]
constexpr long W_XC   = W_XS2 + (long)GB*GK2*GD;      // [B,256]
constexpr long W_H1   = W_XC  + (long)GB*256;         // [B,512]
constexpr long W_H2   = W_H1  + (long)GB*512;         // [B,1024]
constexpr long W_DINV = W_H2  + (long)GB*1024;        // [B,N]
constexpr long W_PART = W_DINV+ (long)GB*GN;          // 256 partial maxima
constexpr long W_GMAX = W_PART+ 256;
constexpr long W_IDX1 = W_GMAX+ 16;                   // ints
constexpr long W_IDX2 = W_IDX1+ (long)GB*GK1;

// -------------------- deterministic hash RNG --------------------
__device__ __forceinline__ unsigned hash_u32(unsigned x) {
  x ^= x >> 17; x *= 0xed5ad4bbu;
  x ^= x >> 11; x *= 0xac4c1b51u;
  x ^= x >> 15; x *= 0x31848babu;
  x ^= x >> 14; return x;
}
__device__ __forceinline__ float u01(unsigned h) {
  return ((float)h + 0.5f) * 2.3283064365386963e-10f; // (0,1)
}
__device__ __forceinline__ float gaussian(unsigned idx) {
  float u1 = u01(hash_u32(idx * 2u + 0x9e3779b9u));
  float u2 = u01(hash_u32(idx * 2u + 0x85ebca6bu));
  return sqrtf(-2.0f * logf(u1)) * cosf(6.2831853071795864f * u2);
}

// -------------------- WMMA fp32 batched strided GEMM --------------------
// C[b](MxN) = epi( A[b](MxK) @ B[b](KxN) + bias ), C row-major contiguous (ldc=N).
// One wave computes a 32x32 C block (2x2 tiles of 16x16) via V_WMMA_F32_16X16X4_F32.
// K must be a multiple of 4 (true for every GEMM in this model).
// Out-of-range rows/cols: load addresses are CLAMPED (A row m only feeds C row m,
// B col n only feeds C col n, so garbage lands exclusively in discarded outputs);
// bounds are enforced only at the store. The K-loop is branch-free.
// EPI: 0=none, 1=+bias, 2=+bias,ReLU.  AK1: sAk==1.  BK1: sBk==1.
template <int EPI, bool AK1, bool BK1>
__global__ void gemm_wmma_f32(const float* __restrict__ A,
                              const float* __restrict__ Bm,
                              const float* __restrict__ bias,
                              float* __restrict__ C,
                              int M, int Nn, int K,
                              long sAb, long sAm, long sAk,
                              long sBb, long sBk, long sBn,
                              long sCb) {
  const int wave = threadIdx.x >> 5;     // 4 waves / block (blockDim=128)
  const int lane = threadIdx.x & 31;
  const int tilesN = (Nn + 31) >> 5;
  const int tilesM = (M + 31) >> 5;
  const int tileId = blockIdx.x * (blockDim.x >> 5) + wave;
  if (tileId >= tilesM * tilesN) return;          // wave-uniform: EXEC stays full
  const int tm = tileId / tilesN, tn = tileId % tilesN;
  const int b = blockIdx.y;
  const int row0 = tm << 5, col0 = tn << 5;
  const int half = lane >> 4, lr = lane & 15;

  // clamped per-lane row/col indices (garbage only reaches discarded outputs)
  const int r0 = min(row0 + lr,      M - 1);
  const int r1 = min(row0 + 16 + lr, M - 1);
  const int c0 = min(col0 + lr,      Nn - 1);
  const int c1 = min(col0 + 16 + lr, Nn - 1);

  const float* Ab = A  + (long)b * sAb;
  const float* Bb = Bm + (long)b * sBb;
  const float* pa0 = Ab + (long)r0 * sAm + (long)(2 * half) * sAk;
  const float* pa1 = Ab + (long)r1 * sAm + (long)(2 * half) * sAk;
  const float* pb0 = Bb + (long)(2 * half) * sBk + (long)c0 * sBn;
  const float* pb1 = Bb + (long)(2 * half) * sBk + (long)c1 * sBn;
  const long dA = 4 * sAk, dB = 4 * sBk;

  v8f acc00 = {}, acc01 = {}, acc10 = {}, acc11 = {};
#pragma unroll 2
  for (int k0 = 0; k0 < K; k0 += 4) {
    v2f af0, af1, bf0, bf1;
    if constexpr (AK1) {                // contiguous K pair -> global_load_b64
      af0 = *(const v2f*)pa0;  af1 = *(const v2f*)pa1;
    } else {
      af0[0] = pa0[0]; af0[1] = pa0[sAk];
      af1[0] = pa1[0]; af1[1] = pa1[sAk];
    }
    if constexpr (BK1) {
      bf0 = *(const v2f*)pb0;  bf1 = *(const v2f*)pb1;
    } else {
      bf0[0] = pb0[0]; bf0[1] = pb0[sBk];
      bf1[0] = pb1[0]; bf1[1] = pb1[sBk];
    }
    // 4 independent WMMAs (distinct D) -> no D->A/B hazard NOPs required
    acc00 = __builtin_amdgcn_wmma_f32_16x16x4_f32(false, af0, false, bf0, (short)0, acc00, false, false);
    acc01 = __builtin_amdgcn_wmma_f32_16x16x4_f32(false, af0, false, bf1, (short)0, acc01, false, false);
    acc10 = __builtin_amdgcn_wmma_f32_16x16x4_f32(false, af1, false, bf0, (short)0, acc10, false, false);
    acc11 = __builtin_amdgcn_wmma_f32_16x16x4_f32(false, af1, false, bf1, (short)0, acc11, false, false);
    pa0 += dA; pa1 += dA; pb0 += dB; pb1 += dB;
  }

  float* Cb = C + (long)b * sCb;
  auto store_tile = [&](const v8f& acc, int rb, int cb) {
#pragma unroll
    for (int v = 0; v < 8; ++v) {
      const int r = rb + v + 8 * half;            // C layout: M=v+8*half, N=lr
      const int c = cb + lr;
      if (r < M && c < Nn) {
        float xv = acc[v];
        if (EPI >= 1) xv += bias[c];
        if (EPI == 2) xv = fmaxf(xv, 0.0f);
        Cb[(long)r * Nn + c] = xv;
      }
    }
  };
  store_tile(acc00, row0,      col0);
  store_tile(acc01, row0,      col0 + 16);
  store_tile(acc10, row0 + 16, col0);
  store_tile(acc11, row0 + 16, col0 + 16);
}

// -------------------- elementwise / reduction kernels --------------------
__global__ void k_binarize(const float* __restrict__ adj, float* __restrict__ aorg, long total) {
  for (long t = blockIdx.x * (long)blockDim.x + threadIdx.x; t < total;
       t += (long)gridDim.x * blockDim.x)
    aorg[t] = adj[t] > 0.0f ? 1.0f : 0.0f;
}

__global__ void k_deg(const float* __restrict__ aorg, float* __restrict__ dinv) {
  long t = blockIdx.x * (long)blockDim.x + threadIdx.x;
  if (t >= (long)GB * GN) return;
  long b = t / GN; int i = (int)(t % GN);
  float deg = 1.0f;                                // self loop
  const float* base = aorg + b * NN;               // symmetric: col-sum == row-sum (coalesced)
  for (int j = 0; j < GN; ++j) deg += base[(long)j * GN + i];
  dinv[t] = rsqrtf(deg);
}

__global__ void k_an(const float* __restrict__ aorg, const float* __restrict__ dinv,
                     float* __restrict__ An) {
  for (long t = blockIdx.x * (long)blockDim.x + threadIdx.x; t < BNN;
       t += (long)gridDim.x * blockDim.x) {
    long b = t / NN; long rem = t % NN;
    int i = (int)(rem / GN), j = (int)(rem % GN);
    float a = aorg[t] + (i == j ? 1.0f : 0.0f);
    An[t] = dinv[b * GN + i] * a * dinv[b * GN + j];
  }
}

__global__ void k_sample_z(float* __restrict__ z /*holds m*/, const float* __restrict__ s,
                           long total) {
  for (long t = blockIdx.x * (long)blockDim.x + threadIdx.x; t < total;
       t += (long)gridDim.x * blockDim.x)
    z[t] = z[t] + gaussian((unsigned)t ^ 0x12345u) * expf(s[t]);
}

__global__ void k_maxred1(const float* __restrict__ in, float* __restrict__ part, long total) {
  __shared__ float sm[256];
  float m = -3.4e38f;
  for (long t = blockIdx.x * (long)blockDim.x + threadIdx.x; t < total;
       t += (long)gridDim.x * blockDim.x)
    m = fmaxf(m, in[t]);
  sm[threadIdx.x] = m; __syncthreads();
  for (int o = 128; o > 0; o >>= 1) {
    if (threadIdx.x < o) sm[threadIdx.x] = fmaxf(sm[threadIdx.x], sm[threadIdx.x + o]);
    __syncthreads();
  }
  if (threadIdx.x == 0) part[blockIdx.x] = sm[0];
}

__global__ void k_maxred2(const float* __restrict__ part, float* __restrict__ gmax, int n) {
  __shared__ float sm[256];
  sm[threadIdx.x] = (threadIdx.x < n) ? part[threadIdx.x] : -3.4e38f;
  __syncthreads();
  for (int o = 128; o > 0; o >>= 1) {
    if (threadIdx.x < o) sm[threadIdx.x] = fmaxf(sm[threadIdx.x], sm[threadIdx.x + o]);
    __syncthreads();
  }
  if (threadIdx.x == 0) *gmax = sm[0];
}

// straight-through relaxed Bernoulli; symmetric fill from upper triangle
__global__ void k_bernoulli(const float* __restrict__ alog, const float* __restrict__ aorg,
                            const float* __restrict__ gmax, float* __restrict__ adj_s) {
  const float inv_maxp = 1.0f + expf(-*gmax);      // 1/sigmoid(gmax)
  for (long t = blockIdx.x * (long)blockDim.x + threadIdx.x; t < BNN;
       t += (long)gridDim.x * blockDim.x) {
    long b = t / NN; long rem = t % NN;
    int i = (int)(rem / GN), j = (int)(rem % GN);
    if (i > j) continue;
    float ps = 1.0f / (1.0f + expf(-alog[t]));
    float pr = 0.5f * (ps * inv_maxp) + 0.5f * aorg[t];
    float logit = logf(pr) - log1pf(-pr);
    float u = 1e-6f + (1.0f - 2e-6f) * u01(hash_u32((unsigned)t ^ 0xA5A5A5A5u));
    float noise = logf(u) - log1pf(-u);
    float soft = 1.0f / (1.0f + expf(-(logit + noise)));   // TEMP=1
    float hard = rintf(soft);                              // forward value of ST estimator
    adj_s[(b * GN + i) * GN + j] = hard;
    adj_s[(b * GN + j) * GN + i] = hard;
  }
}

// TopK pooling: block per batch, iterative argmax (stable: smaller index on ties)
__global__ void k_topk(const float* __restrict__ X, const float* __restrict__ p,
                       float* __restrict__ sig_out, int* __restrict__ idx_out,
                       float* __restrict__ Xs, int n, int k) {
  __shared__ float sc[256];
  __shared__ float rv[256];
  __shared__ int   ri[256];
  __shared__ float pnorm, pickV;
  __shared__ int   pickI;
  const int b = blockIdx.x, t = threadIdx.x;
  if (t == 0) {
    float acc = 0.0f;
    for (int j = 0; j < GD; ++j) acc += p[j] * p[j];
    pnorm = sqrtf(acc);
  }
  __syncthreads();
  if (t < n) {
    float d = 0.0f;
    const float* xr = X + ((long)b * n + t) * GD;
    for (int j = 0; j < GD; ++j) d += xr[j] * p[j];
    sc[t] = tanhf(d / pnorm);
  } else {
    sc[t] = -3.4e38f;
  }
  __syncthreads();
  for (int r = 0; r < k; ++r) {
    rv[t] = sc[t]; ri[t] = t; __syncthreads();
    for (int o = 128; o > 0; o >>= 1) {
      if (t < o) {
        float v2 = rv[t + o]; int i2 = ri[t + o];
        if (v2 > rv[t] || (v2 == rv[t] && i2 < ri[t])) { rv[t] = v2; ri[t] = i2; }
      }
      __syncthreads();
    }
    if (t == 0) {
      pickI = ri[0]; pickV = rv[0];
      idx_out[b * k + r] = pickI;
      sig_out[(long)b * k + r] = 1.0f / (1.0f + expf(-pickV));
      sc[pickI] = -3.4e38f;
    }
    __syncthreads();
    if (t < GD)
      Xs[((long)b * k + r) * GD + t] = X[((long)b * n + pickI) * GD + t] * pickV;
    __syncthreads();
  }
}

__global__ void k_gatherA(const float* __restrict__ A, const int* __restrict__ idx,
                          float* __restrict__ As, int n, int k) {
  long total = (long)GB * k * k;
  for (long t = blockIdx.x * (long)blockDim.x + threadIdx.x; t < total;
       t += (long)gridDim.x * blockDim.x) {
    long b = t / ((long)k * k); long rem = t % ((long)k * k);
    int i = (int)(rem / k), j = (int)(rem % k);
    int ri = idx[b * k + i], rj = idx[b * k + j];
    As[t] = A[(b * n + ri) * (long)n + rj];
  }
}

__global__ void k_poolcat(const float* __restrict__ Xs, float* __restrict__ xc,
                          int k, int off) {
  int b = blockIdx.x, d = threadIdx.x;   // 64 threads
  float mx = -3.4e38f, sm = 0.0f;
  for (int r = 0; r < k; ++r) {
    float v = Xs[((long)b * k + r) * GD + d];
    mx = fmaxf(mx, v); sm += v;
  }
  xc[(long)b * 256 + off + d]      = mx;
  xc[(long)b * 256 + off + GD + d] = sm / (float)k;
}

__global__ void k_bn(float* __restrict__ x, const float* __restrict__ g,
                     const float* __restrict__ be, long total, int F) {
  const float invs = rsqrtf(1.0f + 1e-5f);
  for (long t = blockIdx.x * (long)blockDim.x + threadIdx.x; t < total;
       t += (long)gridDim.x * blockDim.x) {
    int f = (int)(t % F);
    x[t] = g[f] * (x[t] * invs) + be[f];
  }
}

__global__ void k_head(const float* __restrict__ H2, const float* __restrict__ Wf3,
                       const float* __restrict__ bf3, float* __restrict__ out) {
  __shared__ float r0[256], r1[256];
  int b = blockIdx.x, t = threadIdx.x;
  float a0 = 0.0f, a1 = 0.0f;
  for (int j = t; j < 1024; j += 256) {
    float h = H2[(long)b * 1024 + j];
    a0 += h * Wf3[j * 2 + 0];
    a1 += h * Wf3[j * 2 + 1];
  }
  r0[t] = a0; r1[t] = a1; __syncthreads();
  for (int o = 128; o > 0; o >>= 1) {
    if (t < o) { r0[t] += r0[t + o]; r1[t] += r1[t + o]; }
    __syncthreads();
  }
  if (t == 0) {
    float l0 = r0[0] + bf3[0], l1 = r1[0] + bf3[1];
    float m = fmaxf(l0, l1);
    float lse = m + logf(expf(l0 - m) + expf(l1 - m));
    out[b * 2 + 0] = l0 - lse;
    out[b * 2 + 1] = l1 - lse;
  }
}

// -------------------- host-side GEMM launcher --------------------
static void launch_gemm(int epi, const float* A, const float* Bm, const float* bias,
                        float* C, int batch, int M, int Nn, int K,
                        long sAb, long sAm, long sAk,
                        long sBb, long sBk, long sBn, long sCb, hipStream_t st) {
  int tiles = ((M + 31) / 32) * ((Nn + 31) / 32);
  dim3 grid((tiles + 3) / 4, batch), block(128);
  const bool ak1 = (sAk == 1), bk1 = (sBk == 1);
#define GO(E, A1, B1) gemm_wmma_f32<E, A1, B1><<<grid, block, 0, st>>>( \
    A, Bm, bias, C, M, Nn, K, sAb, sAm, sAk, sBb, sBk, sBn, sCb)
  switch (epi) {
    case 0:
      if (ak1 && bk1)      GO(0, true,  true);
      else if (ak1)        GO(0, true,  false);
      else if (bk1)        GO(0, false, true);
      else                 GO(0, false, false);
      break;
    case 1:
      if (ak1 && bk1)      GO(1, true,  true);
      else if (ak1)        GO(1, true,  false);
      else if (bk1)        GO(1, false, true);
      else                 GO(1, false, false);
      break;
    default:
      if (ak1 && bk1)      GO(2, true,  true);
      else if (ak1)        GO(2, true,  false);
      else if (bk1)        GO(2, false, true);
      else                 GO(2, false, false);
      break;
  }
#undef GO
}

} // namespace

extern "C" void kernel_launch(void* const* d_in, const int* in_sizes, int n_in,
                              void* d_out, int out_size, void* d_ws, size_t ws_size,
                              hipStream_t stream) {
  (void)in_sizes; (void)n_in; (void)out_size; (void)ws_size;
  const float* x   = (const float*)d_in[0];
  const float* adj = (const float*)d_in[1];
  const float* Wb  = (const float*)d_in[2];
  const float* bb  = (const float*)d_in[3];
  const float* Wm  = (const float*)d_in[4];
  const float* bm  = (const float*)d_in[5];
  const float* Ws  = (const float*)d_in[6];
  const float* bs  = (const float*)d_in[7];
  const float* W1  = (const float*)d_in[8];
  const float* b1  = (const float*)d_in[9];
  const float* p1  = (const float*)d_in[10];
  const float* W2  = (const float*)d_in[11];
  const float* b2  = (const float*)d_in[12];
  const float* p2  = (const float*)d_in[13];
  const float* Wf1 = (const float*)d_in[14];
  const float* bf1 = (const float*)d_in[15];
  const float* g1  = (const float*)d_in[16];
  const float* be1 = (const float*)d_in[17];
  const float* Wf2 = (const float*)d_in[18];
  const float* bf2 = (const float*)d_in[19];
  const float* g2  = (const float*)d_in[20];
  const float* be2 = (const float*)d_in[21];
  const float* Wf3 = (const float*)d_in[22];
  const float* bf3 = (const float*)d_in[23];

  float* out = (float*)d_out;
  float* ws  = (float*)d_ws;
  float* aorg = out + AORG_OFF;
  float* alog = out + ALOG_OFF;

  const dim3 blk(256);
  const int GS = 4096;  // grid-stride block count for big elementwise kernels

  // ---- adjacency normalize ----
  k_binarize<<<GS, blk, 0, stream>>>(adj, aorg, BNN);
  k_deg<<<(GB * GN + 255) / 256, blk, 0, stream>>>(aorg, ws + W_DINV);
  k_an<<<GS, blk, 0, stream>>>(aorg, ws + W_DINV, ws + W_AN);

  // ---- VGAE encoder ----
  // XB = x @ Wb
  launch_gemm(0, x, Wb, nullptr, ws + W_XB, GB, GN, GH, GN,
              NN, GN, 1, 0, GH, 1, (long)GN * GH, stream);
  // h = An @ XB + bb
  launch_gemm(1, ws + W_AN, ws + W_XB, bb, ws + W_H, GB, GN, GH, GN,
              NN, GN, 1, (long)GN * GH, GH, 1, (long)GN * GH, stream);
  // HM = h @ Wm ; m = relu(An @ HM + bm)
  launch_gemm(0, ws + W_H, Wm, nullptr, ws + W_T0, GB, GN, GN, GH,
              (long)GN * GH, GH, 1, 0, GN, 1, NN, stream);
  launch_gemm(2, ws + W_AN, ws + W_T0, bm, ws + W_T1, GB, GN, GN, GN,
              NN, GN, 1, NN, GN, 1, NN, stream);
  // HS = h @ Ws ; s = relu(An @ HS + bs)
  launch_gemm(0, ws + W_H, Ws, nullptr, ws + W_T0, GB, GN, GN, GH,
              (long)GN * GH, GH, 1, 0, GN, 1, NN, stream);
  launch_gemm(2, ws + W_AN, ws + W_T0, bs, ws + W_T2, GB, GN, GN, GN,
              NN, GN, 1, NN, GN, 1, NN, stream);
  // z = m + eps * exp(s)   (in place over m)
  k_sample_z<<<GS, blk, 0, stream>>>(ws + W_T1, ws + W_T2, BNN);
  // adj_logits = z @ z^T  (B transposed via strides: sBk=1, sBn=GN)
  launch_gemm(0, ws + W_T1, ws + W_T1, nullptr, alog, GB, GN, GN, GN,
              NN, GN, 1, NN, 1, GN, NN, stream);

  // ---- relaxed Bernoulli straight-through ----
  k_maxred1<<<256, blk, 0, stream>>>(alog, ws + W_PART, BNN);
  k_maxred2<<<1, blk, 0, stream>>>(ws + W_PART, ws + W_GMAX, 256);
  k_bernoulli<<<GS, blk, 0, stream>>>(alog, aorg, ws + W_GMAX, ws + W_T0);

  // ---- classifier conv1 + pool1 ----
  launch_gemm(0, x, W1, nullptr, ws + W_XW1, GB, GN, GD, GN,
              NN, GN, 1, 0, GD, 1, (long)GN * GD, stream);
  launch_gemm(2, ws + W_T0, ws + W_XW1, b1, ws + W_X1, GB, GN, GD, GN,
              NN, GN, 1, (long)GN * GD, GD, 1, (long)GN * GD, stream);
  k_topk<<<GB, blk, 0, stream>>>(ws + W_X1, p1, out + S1_OFF,
                                 (int*)(ws + W_IDX1), ws + W_XS1, GN, GK1);
  k_gatherA<<<GS, blk, 0, stream>>>(ws + W_T0, (int*)(ws + W_IDX1), ws + W_AS1, GN, GK1);
  k_poolcat<<<GB, dim3(GD), 0, stream>>>(ws + W_XS1, ws + W_XC, GK1, 0);

  // ---- conv2 + pool2 ----
  launch_gemm(0, ws + W_XS1, W2, nullptr, ws + W_XW2, GB, GK1, GD, GD,
              (long)GK1 * GD, GD, 1, 0, GD, 1, (long)GK1 * GD, stream);
  launch_gemm(2, ws + W_AS1, ws + W_XW2, b2, ws + W_X2, GB, GK1, GD, GK1,
              (long)GK1 * GK1, GK1, 1, (long)GK1 * GD, GD, 1, (long)GK1 * GD, stream);
  k_topk<<<GB, blk, 0, stream>>>(ws + W_X2, p2, out + S2_OFF,
                                 (int*)(ws + W_IDX2), ws + W_XS2, GK1, GK2);
  k_poolcat<<<GB, dim3(GD), 0, stream>>>(ws + W_XS2, ws + W_XC, GK2, 128);

  // ---- MLP head (single "batch", M = GB rows) ----
  launch_gemm(2, ws + W_XC, Wf1, bf1, ws + W_H1, 1, GB, 512, 256,
              0, 256, 1, 0, 512, 1, 0, stream);
  k_bn<<<(GB * 512 + 255) / 256, blk, 0, stream>>>(ws + W_H1, g1, be1, (long)GB * 512, 512);
  launch_gemm(2, ws + W_H1, Wf2, bf2, ws + W_H2, 1, GB, 1024, 512,
              0, 512, 1, 0, 1024, 1, 0, stream);
  k_bn<<<(GB * 1024 + 255) / 256, blk, 0, stream>>>(ws + W_H2, g2, be2, (long)GB * 1024, 1024);
  k_head<<<GB, blk, 0, stream>>>(ws + W_H2, Wf3, bf3, out + XY_OFF);
}